// PointsFusion2_64201171140765
// MI455X (gfx1250) — compile-verified
//
#include <hip/hip_runtime.h>
#include <math.h>

typedef float v2f __attribute__((ext_vector_type(2)));
typedef float v8f __attribute__((ext_vector_type(8)));
typedef unsigned int v4u __attribute__((ext_vector_type(4)));
typedef int v8i_ __attribute__((ext_vector_type(8)));
typedef int v4i_ __attribute__((ext_vector_type(4)));

#define BATCH 8
#define NPTS  4096
#define MPTS  4096
#define KNBR  32
#define C1    32
#define C2    64
#define KCAP  24
#define EPSV  1e-5f
#define CHUNK 1024   /* reference points staged into LDS per TDM transfer */

// workspace layout (float indices); header ints live in bytes [0,64)
#define WSF_Q4  64
#define WSF_P0  (WSF_Q4 + BATCH*NPTS*4)
#define WSF_P1  (WSF_P0 + BATCH*MPTS*4)
#define WSF_GRP (WSF_P1 + BATCH*MPTS*4)
#define WSF_ST  (WSF_GRP + BATCH*3*NPTS*KNBR)
// stats region (floats, relative to WSF_ST):
//   [0..63]    gn1 raw sum/ss  : b*8  + g*2 + {0,1}   (4 groups)
//   [64..127]  gn1 mean/rstd   : b*8  + g*2 + {0,1}
//   [128..255] gn2 raw sum/ss  : b*16 + g*2 + {0,1}   (8 groups)
//   [256..383] gn2 mean/rstd   : b*16 + g*2 + {0,1}

#if __has_builtin(__builtin_amdgcn_tensor_load_to_lds)
#define USE_TDM 1
#else
#define USE_TDM 0
#endif

// ---------------------------------------------------------------- prep ----
__global__ void __launch_bounds__(256)
k_prep(const float* __restrict__ p0, const float* __restrict__ p1,
       const float* __restrict__ wt, const int* __restrict__ perm,
       float* __restrict__ wsf, int* __restrict__ hdr) {
  const int b = blockIdx.y;
  const int j = blockIdx.x * 256 + threadIdx.x;   // [0,4096)
  const float w = wt[b * 2 + 0];
  const int N0 = (int)(NPTS * w);
  const int k0 = (int)(KNBR * w);
  if (j == 0) hdr[b] = k0;
  if (b == 0 && j < 384) wsf[WSF_ST + j] = 0.0f;  // zero GN stat accumulators

  // build query point j of newp (gather through perm with data-dependent split)
  int src; const float* ps;
  if (j < N0) { src = perm[(b * 2 + 0) * NPTS + j];        ps = p0; }
  else        { src = perm[(b * 2 + 1) * NPTS + (j - N0)]; ps = p1; }
  const float qx = ps[(b * 3 + 0) * NPTS + src];
  const float qy = ps[(b * 3 + 1) * NPTS + src];
  const float qz = ps[(b * 3 + 2) * NPTS + src];
  ((float4*)(wsf + WSF_Q4))[b * NPTS + j] = make_float4(qx, qy, qz, 1.0f);

  // prepped reference points: [-2x,-2y,-2z,|p|^2]  (score = |p|^2 - 2 q.p)
  {
    const float ax = p0[(b * 3 + 0) * MPTS + j];
    const float ay = p0[(b * 3 + 1) * MPTS + j];
    const float az = p0[(b * 3 + 2) * MPTS + j];
    ((float4*)(wsf + WSF_P0))[b * MPTS + j] =
        make_float4(-2.f * ax, -2.f * ay, -2.f * az, ax * ax + ay * ay + az * az);
  }
  {
    const float ax = p1[(b * 3 + 0) * MPTS + j];
    const float ay = p1[(b * 3 + 1) * MPTS + j];
    const float az = p1[(b * 3 + 2) * MPTS + j];
    ((float4*)(wsf + WSF_P1))[b * MPTS + j] =
        make_float4(-2.f * ax, -2.f * ay, -2.f * az, ax * ax + ay * ay + az * az);
  }
}

// ----------------------------------------------------------------- knn ----
// grid (16, B, 2), block 256.  Each wave32 owns 32 queries; per 16-ref tile
// two V_WMMA_F32_16X16X4_F32 produce the 32x16 score tile.  Reference points
// are staged global->LDS in 16KB chunks by the Tensor Data Mover.
__global__ void __launch_bounds__(256)
k_knn(const float* __restrict__ p0, const float* __restrict__ p1,
      const int* __restrict__ hdr, float* __restrict__ wsf) {
  __shared__ float4 lprep[CHUNK];      // 16 KB staging buffer (TDM destination)
  __shared__ float4 sc4[8][32][5];     // per-wave 32x16 score tiles, 20-float row stride
  const int wave = threadIdx.x >> 5;
  const int lane = threadIdx.x & 31;
  const int b = blockIdx.y;
  const int set = blockIdx.z;
  const int qbase = (blockIdx.x * 8 + wave) * 32;

  const float4* q4   = (const float4*)(wsf + WSF_Q4) + b * NPTS;
  const float4* prep = (const float4*)(wsf + (set ? WSF_P1 : WSF_P0)) + b * MPTS;
  const bool hi = lane >= 16;
  const int  col = lane & 15;
  const int  r8  = hi ? 8 : 0;

  // A operands: 16x4 f32 layout — lanes 0-15 hold K=0,1 ; lanes 16-31 hold K=2,3
  const float4 qa = q4[qbase + col];
  const float4 qb = q4[qbase + 16 + col];
  v2f A0, A1;
  A0[0] = hi ? qa.z : qa.x;  A0[1] = hi ? qa.w : qa.y;
  A1[0] = hi ? qb.z : qb.x;  A1[1] = hi ? qb.w : qb.y;

  float ld[KCAP]; int li[KCAP];
#pragma unroll
  for (int j = 0; j < KCAP; ++j) { ld[j] = 3.0e38f; li[j] = 0; }

  for (int ch = 0; ch < MPTS / CHUNK; ++ch) {
#if USE_TDM
    if (threadIdx.x < 32) {   // wave 0 issues one TDM transfer per chunk
      const unsigned long long ga =
          (unsigned long long)(uintptr_t)(prep + ch * CHUNK);
      v4u g0;
      g0[0] = 1u;                                        // count=1 descriptor
      g0[1] = (unsigned)(uintptr_t)&lprep[0];            // LDS byte offset
      g0[2] = (unsigned)(ga & 0xFFFFFFFFu);              // global_addr[31:0]
      g0[3] = (unsigned)((ga >> 32) & 0x1FFFFFFu) | (2u << 30);  // [56:32] | type=2
      v8i_ g1;
      g1[0] = 0x00020000;      // data_size = 4 bytes
      g1[1] = 0x10000000;      // tensor_dim0 = 4096 (low16 at bit 48 -> d1[31:16])
      g1[2] = 0x00010000;      // tensor_dim0 hi16=0 ; tensor_dim1 = 1
      g1[3] = 0x10000000;      // tensor_dim1 hi16=0 ; tile_dim0 = 4096
      g1[4] = 0x00000001;      // tile_dim1 = 1, tile_dim2 = 0
      g1[5] = 4096;            // tensor_dim0_stride (low 32)
      g1[6] = 0; g1[7] = 0;
      v4i_ gz4 = {0, 0, 0, 0};
      v8i_ gz8 = {0, 0, 0, 0, 0, 0, 0, 0};
      __builtin_amdgcn_tensor_load_to_lds(g0, g1, gz4, gz4, gz8, 0);
      __builtin_amdgcn_s_wait_tensorcnt(0);
    }
#else
    for (int i = threadIdx.x; i < CHUNK; i += 256)
      lprep[i] = prep[ch * CHUNK + i];
#endif
    __syncthreads();

    for (int t = 0; t < CHUNK / 16; ++t) {
      const float4 pv = lprep[t * 16 + col];
      v2f Bv; Bv[0] = hi ? pv.z : pv.x;  Bv[1] = hi ? pv.w : pv.y;
      v8f acc0 = {}; v8f acc1 = {};
      acc0 = __builtin_amdgcn_wmma_f32_16x16x4_f32(false, A0, false, Bv, (short)0, acc0, false, false);
      acc1 = __builtin_amdgcn_wmma_f32_16x16x4_f32(false, A1, false, Bv, (short)0, acc1, false, false);
#pragma unroll
      for (int r = 0; r < 8; ++r) {
        ((float*)&sc4[wave][r + r8][0])[col]      = acc0[r];
        ((float*)&sc4[wave][16 + r + r8][0])[col] = acc1[r];
      }
      // per-wave LDS region; DS ops are in-order within a wave -> no s_barrier
      __builtin_amdgcn_wave_barrier();
      asm volatile("" ::: "memory");

      float cs[16];
#pragma unroll
      for (int i2 = 0; i2 < 4; ++i2) {   // 4x ds_load_b128: this lane's score row
        const float4 v = sc4[wave][lane][i2];
        cs[i2 * 4 + 0] = v.x; cs[i2 * 4 + 1] = v.y;
        cs[i2 * 4 + 2] = v.z; cs[i2 * 4 + 3] = v.w;
      }
      const int refBase = ch * CHUNK + t * 16;
#pragma unroll
      for (int c = 0; c < 16; ++c) {
        const float cd = cs[c];
        if (cd < ld[KCAP - 1]) {         // tail insert + in-place CAS chain
          ld[KCAP - 1] = cd; li[KCAP - 1] = refBase + c;
#pragma unroll
          for (int j = KCAP - 1; j > 0; --j) {
            const bool sw = ld[j] < ld[j - 1];
            const float fa = ld[j - 1], fb = ld[j];
            const int   ia = li[j - 1], ib = li[j];
            ld[j - 1] = sw ? fb : fa;  ld[j] = sw ? fa : fb;
            li[j - 1] = sw ? ib : ia;  li[j] = sw ? ia : ib;
          }
        }
      }
      __builtin_amdgcn_wave_barrier();
      asm volatile("" ::: "memory");
    }
    __syncthreads();   // chunk buffer about to be overwritten
  }

  const int k0 = hdr[b];
  const int kk    = set ? (KNBR - k0) : k0;
  const int slot0 = set ? k0 : 0;
  const float* pts = set ? p1 : p0;
  const int q = qbase + lane;
  float* grp = wsf + WSF_GRP;
#pragma unroll
  for (int j = 0; j < KCAP; ++j) {
    if (j < kk) {
      const int idx = li[j];
#pragma unroll
      for (int c = 0; c < 3; ++c)
        grp[((b * 3 + c) * NPTS + q) * KNBR + slot0 + j] = pts[(b * 3 + c) * MPTS + idx];
    }
  }
}

// ------------------------------------------------------- feat recompute ---
__device__ __forceinline__ void load_feat(const float* wsf, int b, int n, int slot,
                                          float& rx, float& ry, float& rz, float& ds,
                                          float& nx, float& ny, float& nz) {
  const float* grp = wsf + WSF_GRP;
  nx = grp[((b * 3 + 0) * NPTS + n) * KNBR + slot];
  ny = grp[((b * 3 + 1) * NPTS + n) * KNBR + slot];
  nz = grp[((b * 3 + 2) * NPTS + n) * KNBR + slot];
  const float4 q = ((const float4*)(wsf + WSF_Q4))[b * NPTS + n];
  rx = nx - q.x; ry = ny - q.y; rz = nz - q.z;
  ds = sqrtf(rx * rx + ry * ry + rz * rz);
}

// --------------------------------------------------------------- stats1 ---
__global__ void __launch_bounds__(256)
k_stats1(const float* __restrict__ w1, const float* __restrict__ b1,
         float* __restrict__ wsf) {
  __shared__ float w1s[C1 * 4], b1s[C1], rs[4], rss[4];
  const int tid = threadIdx.x;
  const int b = blockIdx.y;
  if (tid < C1 * 4) w1s[tid] = w1[tid];
  if (tid < C1) b1s[tid] = b1[tid];
  if (tid < 4) { rs[tid] = 0.f; rss[tid] = 0.f; }
  __syncthreads();

  const int t = blockIdx.x * 256 + tid;  // [0, N*K)
  const int n = t >> 5, slot = t & 31;
  float rx, ry, rz, ds, nx, ny, nz;
  load_feat(wsf, b, n, slot, rx, ry, rz, ds, nx, ny, nz);

  float gs[4] = {0, 0, 0, 0}, gq[4] = {0, 0, 0, 0};
#pragma unroll
  for (int o = 0; o < C1; ++o) {
    const float x = b1s[o] + w1s[o*4]*rx + w1s[o*4+1]*ry + w1s[o*4+2]*rz + w1s[o*4+3]*ds;
    gs[o >> 3] += x; gq[o >> 3] += x * x;
  }
#pragma unroll
  for (int g = 0; g < 4; ++g) { atomicAdd(&rs[g], gs[g]); atomicAdd(&rss[g], gq[g]); }
  __syncthreads();
  if (tid < 4) {
    atomicAdd(&wsf[WSF_ST + b * 8 + tid * 2 + 0], rs[tid]);
    atomicAdd(&wsf[WSF_ST + b * 8 + tid * 2 + 1], rss[tid]);
  }
}

// ------------------------------------------------------------- finalize ---
__global__ void k_fin(float* __restrict__ wsf, int rawOff, int outOff, int nent) {
  const int i = threadIdx.x;
  if (i < nent) {
    const float cnt = 8.0f * NPTS * KNBR;  // channels-per-group * N * K
    const float s  = wsf[WSF_ST + rawOff + i * 2];
    const float ss = wsf[WSF_ST + rawOff + i * 2 + 1];
    const float mu = s / cnt;
    const float var = ss / cnt - mu * mu;
    wsf[WSF_ST + outOff + i * 2]     = mu;
    wsf[WSF_ST + outOff + i * 2 + 1] = rsqrtf(var + EPSV);
  }
}

// --------------------------------------------------------------- stats2 ---
__global__ void __launch_bounds__(256)
k_stats2(const float* __restrict__ w1, const float* __restrict__ b1,
         const float* __restrict__ g1w, const float* __restrict__ g1b,
         const float* __restrict__ w2, const float* __restrict__ b2,
         float* __restrict__ wsf) {
  __shared__ float w1s[C1 * 4], b1s[C1], g1ws[C1], g1bs[C1];
  __shared__ float w2s[C2 * C1], b2s[C2];
  __shared__ float m1[4], r1[4], rs[8], rss[8];
  const int tid = threadIdx.x;
  const int b = blockIdx.y;
  for (int i = tid; i < C2 * C1; i += 256) w2s[i] = w2[i];
  if (tid < C1 * 4) w1s[tid] = w1[tid];
  if (tid < C1) { b1s[tid] = b1[tid]; g1ws[tid] = g1w[tid]; g1bs[tid] = g1b[tid]; }
  if (tid < C2) b2s[tid] = b2[tid];
  if (tid < 4) { m1[tid] = wsf[WSF_ST + 64 + b * 8 + tid * 2];
                 r1[tid] = wsf[WSF_ST + 64 + b * 8 + tid * 2 + 1]; }
  if (tid < 8) { rs[tid] = 0.f; rss[tid] = 0.f; }
  __syncthreads();

  const int t = blockIdx.x * 256 + tid;
  const int n = t >> 5, slot = t & 31;
  float rx, ry, rz, ds, nx, ny, nz;
  load_feat(wsf, b, n, slot, rx, ry, rz, ds, nx, ny, nz);

  float x1n[C1];
#pragma unroll
  for (int o = 0; o < C1; ++o) {
    float x = b1s[o] + w1s[o*4]*rx + w1s[o*4+1]*ry + w1s[o*4+2]*rz + w1s[o*4+3]*ds;
    const int g = o >> 3;
    x = (x - m1[g]) * r1[g] * g1ws[o] + g1bs[o];
    x1n[o] = fmaxf(x, 0.f);
  }
#pragma unroll
  for (int g = 0; g < 8; ++g) {
    float s = 0.f, sq = 0.f;
#pragma unroll
    for (int oo = 0; oo < 8; ++oo) {
      const int o2 = g * 8 + oo;
      float a = b2s[o2];
#pragma unroll
      for (int c = 0; c < C1; ++c) a += w2s[o2 * C1 + c] * x1n[c];
      s += a; sq += a * a;
    }
    atomicAdd(&rs[g], s); atomicAdd(&rss[g], sq);
  }
  __syncthreads();
  if (tid < 8) {
    atomicAdd(&wsf[WSF_ST + 128 + b * 16 + tid * 2 + 0], rs[tid]);
    atomicAdd(&wsf[WSF_ST + 128 + b * 16 + tid * 2 + 1], rss[tid]);
  }
}

// ---------------------------------------------------------------- final ---
// grid (512, B), block 256: one wave per (b,n); lane = neighbor slot.
__global__ void __launch_bounds__(256)
k_final(const float* __restrict__ w1, const float* __restrict__ b1,
        const float* __restrict__ g1w, const float* __restrict__ g1b,
        const float* __restrict__ w2, const float* __restrict__ b2,
        const float* __restrict__ g2w, const float* __restrict__ g2b,
        const float* __restrict__ wsf, float* __restrict__ out) {
  __shared__ float w1s[C1 * 4], b1s[C1], g1ws[C1], g1bs[C1];
  __shared__ float w2s[C2 * C1], b2s[C2], g2ws[C2], g2bs[C2];
  __shared__ float m1[4], r1[4], m2[8], r2[8];
  const int tid = threadIdx.x;
  const int b = blockIdx.y;
  for (int i = tid; i < C2 * C1; i += 256) w2s[i] = w2[i];
  if (tid < C1 * 4) w1s[tid] = w1[tid];
  if (tid < C1) { b1s[tid] = b1[tid]; g1ws[tid] = g1w[tid]; g1bs[tid] = g1b[tid]; }
  if (tid < C2) { b2s[tid] = b2[tid]; g2ws[tid] = g2w[tid]; g2bs[tid] = g2b[tid]; }
  if (tid < 4) { m1[tid] = wsf[WSF_ST + 64 + b * 8 + tid * 2];
                 r1[tid] = wsf[WSF_ST + 64 + b * 8 + tid * 2 + 1]; }
  if (tid < 8) { m2[tid] = wsf[WSF_ST + 256 + b * 16 + tid * 2];
                 r2[tid] = wsf[WSF_ST + 256 + b * 16 + tid * 2 + 1]; }
  __syncthreads();

  const int wave = tid >> 5, lane = tid & 31;
  const int n = blockIdx.x * 8 + wave;
  float rx, ry, rz, ds, nx, ny, nz;
  load_feat(wsf, b, n, lane, rx, ry, rz, ds, nx, ny, nz);

  float x1n[C1];
#pragma unroll
  for (int o = 0; o < C1; ++o) {
    float x = b1s[o] + w1s[o*4]*rx + w1s[o*4+1]*ry + w1s[o*4+2]*rz + w1s[o*4+3]*ds;
    const int g = o >> 3;
    x = (x - m1[g]) * r1[g] * g1ws[o] + g1bs[o];
    x1n[o] = fmaxf(x, 0.f);
  }
  float smax = -3.0e38f;
#pragma unroll
  for (int g = 0; g < 8; ++g) {
#pragma unroll
    for (int oo = 0; oo < 8; ++oo) {
      const int o2 = g * 8 + oo;
      float a = b2s[o2];
#pragma unroll
      for (int c = 0; c < C1; ++c) a += w2s[o2 * C1 + c] * x1n[c];
      a = (a - m2[g]) * r2[g] * g2ws[o2] + g2bs[o2];
      a = fmaxf(a, 0.f);
      smax = fmaxf(smax, a);
    }
  }
  // softmax over the 32 neighbor slots (wave32: one slot per lane)
  float m = smax;
  for (int off = 16; off > 0; off >>= 1) m = fmaxf(m, __shfl_xor(m, off, 32));
  const float e = __expf(smax - m);
  float s = e;
  for (int off = 16; off > 0; off >>= 1) s += __shfl_xor(s, off, 32);
  const float wgt = e / s;
  float ox = wgt * nx, oy = wgt * ny, oz = wgt * nz;
  for (int off = 16; off > 0; off >>= 1) {
    ox += __shfl_xor(ox, off, 32);
    oy += __shfl_xor(oy, off, 32);
    oz += __shfl_xor(oz, off, 32);
  }
  if (lane == 0) {
    out[(b * 3 + 0) * NPTS + n] = ox;
    out[(b * 3 + 1) * NPTS + n] = oy;
    out[(b * 3 + 2) * NPTS + n] = oz;
  }
}

// ---------------------------------------------------------------- launch --
extern "C" void kernel_launch(void* const* d_in, const int* in_sizes, int n_in,
                              void* d_out, int out_size, void* d_ws, size_t ws_size,
                              hipStream_t stream) {
  (void)in_sizes; (void)n_in; (void)out_size; (void)ws_size;
  const float* p0   = (const float*)d_in[0];
  const float* p1   = (const float*)d_in[1];
  const float* wt   = (const float*)d_in[3];
  const int*   perm = (const int*)d_in[4];
  const float* w1   = (const float*)d_in[5];
  const float* b1   = (const float*)d_in[6];
  const float* g1w  = (const float*)d_in[7];
  const float* g1b  = (const float*)d_in[8];
  const float* w2   = (const float*)d_in[9];
  const float* b2   = (const float*)d_in[10];
  const float* g2w  = (const float*)d_in[11];
  const float* g2b  = (const float*)d_in[12];
  float* wsf = (float*)d_ws;
  int*   hdr = (int*)d_ws;
  float* out = (float*)d_out;

  k_prep<<<dim3(16, BATCH), 256, 0, stream>>>(p0, p1, wt, perm, wsf, hdr);
  k_knn<<<dim3(16, BATCH, 2), 256, 0, stream>>>(p0, p1, hdr, wsf);
  k_stats1<<<dim3(NPTS * KNBR / 256, BATCH), 256, 0, stream>>>(w1, b1, wsf);
  k_fin<<<dim3(1), 64, 0, stream>>>(wsf, 0, 64, BATCH * 4);
  k_stats2<<<dim3(NPTS * KNBR / 256, BATCH), 256, 0, stream>>>(w1, b1, g1w, g1b, w2, b2, wsf);
  k_fin<<<dim3(1), 64, 0, stream>>>(wsf, 128, 256, BATCH * 8);
  k_final<<<dim3(NPTS / 8, BATCH), 256, 0, stream>>>(w1, b1, g1w, g1b, w2, b2, g2w, g2b, wsf, out);
}